// CrossScanner_42339787604671
// MI455X (gfx1250) — compile-verified
//
#include <hip/hip_runtime.h>
#include <hip/hip_bf16.h>

// CrossScanner (selective-scan / SSM block) for MI455X (gfx1250, wave32).
// Pipeline:
//   k1 proj_kernel      : x_proj GEMMs via V_WMMA_F32_16X16X4_F32 -> dts_raw, Bs, Cs
//   k2 dtproj_kernel    : dt GEMM via WMMA + softplus -> delta
//   k3 scan_chunk_kernel: per-chunk local scan (H) + decay product (P)
//   k4 scan_combine     : serial combine over 64 chunks -> chunk initial states S
//   k5 scan_final_kernel: rescan chunks with correct init state, emit y
// B/C chunk staging uses GLOBAL_LOAD_ASYNC_TO_LDS_B32 (ASYNCcnt) with a
// per-lane LDS-scatter that performs the [n][l] -> [l][n] transpose for free.

typedef __attribute__((ext_vector_type(2))) float v2f;
typedef __attribute__((ext_vector_type(8))) float v8f;

#define B_  2
#define K_  4
#define D_  128
#define L_  4096
#define N_  16
#define R_  8
#define C_  40          // R + 2N channels in x_proj_weight
#define LT_ (L_ / 16)   // 256 column tiles
#define CS_ 64          // scan chunk size
#define NC_ (L_ / CS_)  // 64 chunks

// Async global->LDS copy of one dword per lane (CDNA5 GLOBAL_LOAD_ASYNC_TO_LDS_B32).
// The flat address of an LDS object carries the LDS byte offset in its low 32
// bits (ISA 10.2 aperture mapping), which is exactly the VDST operand.
__device__ __forceinline__ void async_copy_b32(const float* g, float* l)
{
    unsigned lds_addr = (unsigned)(unsigned long long)l;
    unsigned long long gaddr = (unsigned long long)g;
    asm volatile("global_load_async_to_lds_b32 %0, %1, off"
                 :: "v"(lds_addr), "v"(gaddr) : "memory");
}
__device__ __forceinline__ void wait_asynccnt0()
{
    asm volatile("s_wait_asynccnt 0" ::: "memory");
}

// ---------------------------------------------------------------------------
// k1: out[c,l] = sum_d W[k][c][d] * X[b,k][d][l] for the needed channel rows.
// One wave = one 16x16 f32 tile, K=128 in 32 wmma steps of 4.
// wave0: kv rows 0..15 (dts 0..7, Bs 0..7)
// wave1: kv rows 16..31 (Bs 8..15; rows 24..31 discarded)
// wave2: q  rows 24..39 (Cs 0..15)
// ---------------------------------------------------------------------------
__global__ __launch_bounds__(96) void proj_kernel(
    const float* __restrict__ q_x, const float* __restrict__ kv_x,
    const float* __restrict__ Wp,
    float* __restrict__ dts_raw, float* __restrict__ Bs, float* __restrict__ Cs)
{
    const int lt = blockIdx.x % LT_;
    const int bk = blockIdx.x / LT_;
    const int k  = bk % K_;
    const int l0 = lt * 16;
    const int wave = threadIdx.x >> 5;
    const int lane = threadIdx.x & 31;
    const int m  = lane & 15;   // A: M row / B: N col
    const int hi = lane >> 4;   // selects K pair {0,1} vs {2,3}

    const int c0 = (wave < 2) ? (wave * 16) : 24;
    const float* X = ((wave < 2) ? kv_x : q_x) + (size_t)bk * D_ * L_;
    const float* W = Wp + (size_t)k * C_ * D_ + (size_t)(c0 + m) * D_;

    v8f acc = {};
    for (int kk = 0; kk < D_; kk += 4) {
        v2f a, b;
        a.x = W[kk + 2 * hi + 0];
        a.y = W[kk + 2 * hi + 1];
        b.x = X[(size_t)(kk + 2 * hi + 0) * L_ + l0 + m];
        b.y = X[(size_t)(kk + 2 * hi + 1) * L_ + l0 + m];
        acc = __builtin_amdgcn_wmma_f32_16x16x4_f32(
            false, a, false, b, (short)0, acc, false, false);
    }

    const int col = l0 + m;
    #pragma unroll
    for (int i = 0; i < 8; ++i) {
        const int ch = c0 + i + 8 * hi;  // C/D layout: VGPR i -> M = i + 8*hi
        const float v = acc[i];
        if (wave < 2) {
            if (ch < R_)
                dts_raw[((size_t)bk * R_ + ch) * L_ + col] = v;
            else if (ch < R_ + N_)
                Bs[((size_t)bk * N_ + (ch - R_)) * L_ + col] = v;
            // ch >= 24 on the kv path is unused
        } else {
            Cs[((size_t)bk * N_ + (ch - 24)) * L_ + col] = v;
        }
    }
}

// ---------------------------------------------------------------------------
// k2: delta[d,l] = softplus( sum_r dtW[k][d][r]*dts_raw[r][l] + bias[k][d] )
// 8 waves per block, one 16(d)x16(l) tile each; K=8 in 2 wmma steps.
// ---------------------------------------------------------------------------
__global__ __launch_bounds__(256) void dtproj_kernel(
    const float* __restrict__ dtw, const float* __restrict__ dtb,
    const float* __restrict__ dts_raw, float* __restrict__ delta)
{
    const int lt = blockIdx.x % LT_;
    const int bk = blockIdx.x / LT_;
    const int k  = bk % K_;
    const int l0 = lt * 16;
    const int wave = threadIdx.x >> 5;
    const int lane = threadIdx.x & 31;
    const int m  = lane & 15;
    const int hi = lane >> 4;
    const int d0 = wave * 16;

    const float* Arow = dtw + ((size_t)k * D_ + d0 + m) * R_;
    const float* Brow = dts_raw + (size_t)bk * R_ * L_;

    v8f acc = {};
    #pragma unroll
    for (int kk = 0; kk < R_; kk += 4) {
        v2f a, b;
        a.x = Arow[kk + 2 * hi + 0];
        a.y = Arow[kk + 2 * hi + 1];
        b.x = Brow[(size_t)(kk + 2 * hi + 0) * L_ + l0 + m];
        b.y = Brow[(size_t)(kk + 2 * hi + 1) * L_ + l0 + m];
        acc = __builtin_amdgcn_wmma_f32_16x16x4_f32(
            false, a, false, b, (short)0, acc, false, false);
    }

    #pragma unroll
    for (int i = 0; i < 8; ++i) {
        const int d = d0 + i + 8 * hi;
        const float x = acc[i] + dtb[k * D_ + d];
        const float sp = (x > 20.0f) ? x : __logf(1.0f + __expf(x));
        delta[((size_t)bk * D_ + d) * L_ + l0 + m] = sp;
    }
}

// ---------------------------------------------------------------------------
// k3: per-chunk local scan. Block = (b,k,chunk), thread = d.
// H[n] = local state with h_init = 0;  P[n] = prod of dA over the chunk.
// Bs chunk async-scattered into LDS transposed to [l][n].
// ---------------------------------------------------------------------------
__global__ __launch_bounds__(128) void scan_chunk_kernel(
    const float* __restrict__ kv_x, const float* __restrict__ delta,
    const float* __restrict__ Bs, const float* __restrict__ A_logs,
    float* __restrict__ Pws, float* __restrict__ Hws)
{
    __shared__ float sB[CS_ * N_];
    const int c  = blockIdx.x % NC_;
    const int bk = blockIdx.x / NC_;
    const int k  = bk % K_;
    const int d  = threadIdx.x;
    const int lbase = c * CS_;

    const float* gB = Bs + (size_t)bk * N_ * L_ + lbase;
    #pragma unroll
    for (int i = threadIdx.x; i < N_ * CS_; i += 128) {
        const int n = i / CS_, j = i % CS_;
        async_copy_b32(gB + (size_t)n * L_ + j, &sB[j * N_ + n]);
    }
    wait_asynccnt0();
    __syncthreads();

    float Ac[N_];
    #pragma unroll
    for (int n = 0; n < N_; ++n)
        Ac[n] = -__expf(A_logs[(size_t)(k * D_ + d) * N_ + n]);

    float P[N_], H[N_];
    #pragma unroll
    for (int n = 0; n < N_; ++n) { P[n] = 1.0f; H[n] = 0.0f; }

    const float* up = kv_x  + ((size_t)bk * D_ + d) * L_ + lbase;
    const float* dp = delta + ((size_t)bk * D_ + d) * L_ + lbase;

    for (int j4 = 0; j4 < CS_; j4 += 4) {
        const float4 u4 = *(const float4*)(up + j4);
        const float4 d4 = *(const float4*)(dp + j4);
        const float* us = (const float*)&u4;
        const float* ds = (const float*)&d4;
        #pragma unroll
        for (int t = 0; t < 4; ++t) {
            const float dl = ds[t];
            const float du = dl * us[t];
            const int j = j4 + t;
            #pragma unroll
            for (int n = 0; n < N_; ++n) {
                const float dA = __expf(dl * Ac[n]);
                H[n] = dA * H[n] + du * sB[j * N_ + n];
                P[n] *= dA;
            }
        }
    }
    #pragma unroll
    for (int n = 0; n < N_; ++n) {
        const size_t idx = (((size_t)bk * D_ + d) * N_ + n) * NC_ + c;
        Pws[idx] = P[n];
        Hws[idx] = H[n];
    }
}

// ---------------------------------------------------------------------------
// k4: serial combine over NC_ chunks per state chain (16384 chains).
// S[c] = initial state for chunk c;  s_{c+1} = H[c] + P[c]*s_c.
// ---------------------------------------------------------------------------
__global__ __launch_bounds__(256) void scan_combine_kernel(
    const float* __restrict__ Pws, const float* __restrict__ Hws,
    float* __restrict__ Sws)
{
    const size_t sidx = (size_t)blockIdx.x * 256 + threadIdx.x;  // (bk,d,n) flat
    float s = 0.0f;
    for (int c = 0; c < NC_; ++c) {
        Sws[sidx * NC_ + c] = s;
        s = Hws[sidx * NC_ + c] + Pws[sidx * NC_ + c] * s;
    }
}

// ---------------------------------------------------------------------------
// k5: rescan each chunk with its correct initial state, emit y.
// ---------------------------------------------------------------------------
__global__ __launch_bounds__(128) void scan_final_kernel(
    const float* __restrict__ kv_x, const float* __restrict__ delta,
    const float* __restrict__ Bs, const float* __restrict__ Cs,
    const float* __restrict__ A_logs, const float* __restrict__ Ds,
    const float* __restrict__ Sws, float* __restrict__ out)
{
    __shared__ float sB[CS_ * N_];
    __shared__ float sC[CS_ * N_];
    const int c  = blockIdx.x % NC_;
    const int bk = blockIdx.x / NC_;
    const int k  = bk % K_;
    const int d  = threadIdx.x;
    const int lbase = c * CS_;

    const float* gB = Bs + (size_t)bk * N_ * L_ + lbase;
    const float* gC = Cs + (size_t)bk * N_ * L_ + lbase;
    #pragma unroll
    for (int i = threadIdx.x; i < N_ * CS_; i += 128) {
        const int n = i / CS_, j = i % CS_;
        async_copy_b32(gB + (size_t)n * L_ + j, &sB[j * N_ + n]);
        async_copy_b32(gC + (size_t)n * L_ + j, &sC[j * N_ + n]);
    }
    wait_asynccnt0();
    __syncthreads();

    float Ac[N_], h[N_];
    #pragma unroll
    for (int n = 0; n < N_; ++n) {
        Ac[n] = -__expf(A_logs[(size_t)(k * D_ + d) * N_ + n]);
        h[n]  = Sws[(((size_t)bk * D_ + d) * N_ + n) * NC_ + c];
    }
    const float Dsv = Ds[k * D_ + d];

    const float* up = kv_x  + ((size_t)bk * D_ + d) * L_ + lbase;
    const float* dp = delta + ((size_t)bk * D_ + d) * L_ + lbase;
    float* op = out + ((size_t)bk * D_ + d) * L_ + lbase;

    for (int j4 = 0; j4 < CS_; j4 += 4) {
        const float4 u4 = *(const float4*)(up + j4);
        const float4 d4 = *(const float4*)(dp + j4);
        const float* us = (const float*)&u4;
        const float* ds = (const float*)&d4;
        float4 y4;
        float* ys = (float*)&y4;
        #pragma unroll
        for (int t = 0; t < 4; ++t) {
            const float dl = ds[t];
            const float u  = us[t];
            const float du = dl * u;
            const int j = j4 + t;
            float y = 0.0f;
            #pragma unroll
            for (int n = 0; n < N_; ++n) {
                const float dA = __expf(dl * Ac[n]);
                h[n] = dA * h[n] + du * sB[j * N_ + n];
                y += h[n] * sC[j * N_ + n];
            }
            ys[t] = y + u * Dsv;
        }
        *(float4*)(op + j4) = y4;
    }
}

// ---------------------------------------------------------------------------
extern "C" void kernel_launch(void* const* d_in, const int* in_sizes, int n_in,
                              void* d_out, int out_size, void* d_ws, size_t ws_size,
                              hipStream_t stream)
{
    const float* q_x  = (const float*)d_in[0];  // (B,K,D,L)
    const float* kv_x = (const float*)d_in[1];  // (B,K,D,L)
    const float* Wp   = (const float*)d_in[2];  // (K,40,D)
    const float* dtw  = (const float*)d_in[3];  // (K,D,R)
    const float* dtb  = (const float*)d_in[4];  // (K,D)
    const float* Alog = (const float*)d_in[5];  // (K*D,N)
    const float* Dsp  = (const float*)d_in[6];  // (K*D,)
    float* out = (float*)d_out;                 // (B,K,D,L)

    // Workspace layout (floats). Total = 8,650,752 floats = 34.6 MB.
    float* ws = (float*)d_ws;
    float* ws_dts   = ws;                                   // B*K*R*L    = 262144
    float* ws_Bs    = ws_dts + (size_t)B_ * K_ * R_ * L_;   // B*K*N*L    = 524288
    float* ws_Cs    = ws_Bs  + (size_t)B_ * K_ * N_ * L_;   // B*K*N*L    = 524288
    float* ws_delta = ws_Cs  + (size_t)B_ * K_ * N_ * L_;   // B*K*D*L    = 4194304
    float* ws_P     = ws_delta + (size_t)B_ * K_ * D_ * L_; // B*K*D*N*NC = 1048576
    float* ws_H     = ws_P  + (size_t)B_ * K_ * D_ * N_ * NC_;
    float* ws_S     = ws_H  + (size_t)B_ * K_ * D_ * N_ * NC_;

    // k1: projection GEMMs (WMMA). 2048 blocks x 3 waves.
    proj_kernel<<<B_ * K_ * LT_, 96, 0, stream>>>(
        q_x, kv_x, Wp, ws_dts, ws_Bs, ws_Cs);

    // k2: dt projection + softplus (WMMA). 2048 blocks x 8 waves.
    dtproj_kernel<<<B_ * K_ * LT_, 256, 0, stream>>>(
        dtw, dtb, ws_dts, ws_delta);

    // k3: per-chunk local scan. 512 blocks x 128 threads.
    scan_chunk_kernel<<<B_ * K_ * NC_, 128, 0, stream>>>(
        kv_x, ws_delta, ws_Bs, Alog, ws_P, ws_H);

    // k4: cross-chunk combine. 16384 chains.
    scan_combine_kernel<<<(B_ * K_ * D_ * N_) / 256, 256, 0, stream>>>(
        ws_P, ws_H, ws_S);

    // k5: final rescan + output.
    scan_final_kernel<<<B_ * K_ * NC_, 128, 0, stream>>>(
        kv_x, ws_delta, ws_Bs, ws_Cs, Alog, Dsp, ws_S, out);
}